// GAE_56332791054838
// MI455X (gfx1250) — compile-verified
//
#include <hip/hip_runtime.h>
#include <hip/hip_bf16.h>
#include <math.h>

// ---------------------------------------------------------------------------
// GAE forward for MI455X (gfx1250, wave32, WMMA).
//   h1 = x @ W1 + b1                  (fp32 WMMA 16x16x4)
//   h  = spmm(adj, h1)          (gather + global_atomic_add_f32 scatter)
//   z1 = relu(h) @ W2 + b2            (fp32 WMMA, relu fused into A load)
//   z  = spmm(adj, z1)                (scatter)
//   adj_rec = sigmoid(z @ z^T)        (fp32 WMMA; 268MB write-only output
//                                      -> non-temporal stores so the stream
//                                      doesn't sweep the 192MB L2)
// ---------------------------------------------------------------------------

typedef __attribute__((ext_vector_type(2))) float v2f;
typedef __attribute__((ext_vector_type(4))) float v4f;
typedef __attribute__((ext_vector_type(8))) float v8f;

#define N_NODES 8192
#define F_IN    512
#define HID     256
#define EMB     64
#define N_EDGE  262144

__global__ void zero_f32(float* __restrict__ p, int n) {
    int i = blockIdx.x * blockDim.x + threadIdx.x;
    if (i < n) p[i] = 0.0f;
}

// ---------------------------------------------------------------------------
// C[M,NCOLS] = (RELU_A ? relu(A) : A)[M,K] @ B[K,NCOLS] + bias[NCOLS]
// One wave32 computes one 16x16 tile of C using V_WMMA_F32_16X16X4_F32.
//
// fp32 WMMA operand layouts (ISA 7.12.2):
//   A (16x4):  lane L holds M = L%16; VGPR0/1 hold K = 2*(L/16) + {0,1}
//              -> per k-step, lane loads A[row][k + 2*hi .. +1] as one v2f.
//   B (4x16):  lane L holds N = L%16; VGPR0/1 hold K = 2*(L/16) + {0,1}
//              -> two scalar loads strided by NCOLS (each is 2x64B segments,
//                 W panels are L2-resident).
//   C/D:       VGPR r, lane L -> (M = r + 8*(L/16), N = L%16).
// ---------------------------------------------------------------------------
template <int K, int NCOLS, bool RELU_A>
__global__ void gemm_bias_wmma(const float* __restrict__ A,
                               const float* __restrict__ B,
                               const float* __restrict__ bias,
                               float* __restrict__ C) {
    const int lane = threadIdx.x & 31;
    const int wave = (blockIdx.x * blockDim.x + threadIdx.x) >> 5;
    const int tilesN = NCOLS / 16;
    const int tm = wave / tilesN;
    const int tn = wave % tilesN;
    const int l16 = lane & 15;
    const int hi  = lane >> 4;

    const float bv = bias[tn * 16 + l16];
    v8f acc = {bv, bv, bv, bv, bv, bv, bv, bv};

    const float* aRow = A + (size_t)(tm * 16 + l16) * K + 2 * hi;
    const float* bCol = B + (size_t)(2 * hi) * NCOLS + tn * 16 + l16;

#pragma unroll 4
    for (int k = 0; k < K; k += 4) {
        v2f a = *(const v2f*)(aRow + k);
        if (RELU_A) { a.x = fmaxf(a.x, 0.0f); a.y = fmaxf(a.y, 0.0f); }
        v2f b;
        b.x = bCol[(size_t)(k + 0) * NCOLS];
        b.y = bCol[(size_t)(k + 1) * NCOLS];
        acc = __builtin_amdgcn_wmma_f32_16x16x4_f32(
            /*neg_a=*/false, a, /*neg_b=*/false, b,
            /*c_mod=*/(short)0, acc, /*reuse_a=*/false, /*reuse_b=*/false);
    }

    float* cOut = C + (size_t)(tm * 16 + 8 * hi) * NCOLS + tn * 16 + l16;
#pragma unroll
    for (int r = 0; r < 8; ++r) cOut[(size_t)r * NCOLS] = acc[r];
}

// ---------------------------------------------------------------------------
// COO SpMM scatter: out[rows[e], :] += vals[e] * dense[cols[e], :]
// One thread per (edge, 4-feature chunk); dense gather is a b128 load that
// stays L2-resident (dense <= 8MB << 192MB L2); scatter uses HW f32 atomics.
// ---------------------------------------------------------------------------
template <int F>
__global__ void spmm_scatter(const int* __restrict__ rows,
                             const int* __restrict__ cols,
                             const float* __restrict__ vals,
                             const float* __restrict__ dense,
                             float* __restrict__ out) {
    const int chunks = F / 4;
    int tid = blockIdx.x * blockDim.x + threadIdx.x;
    int e = tid / chunks;
    int c = tid - e * chunks;
    if (e >= N_EDGE) return;
    const int   row = rows[e];
    const int   col = cols[e];
    const float v   = vals[e];
    v4f g = *(const v4f*)(dense + (size_t)col * F + c * 4);
    float* dst = out + (size_t)row * F + c * 4;
    unsafeAtomicAdd(dst + 0, v * g.x);
    unsafeAtomicAdd(dst + 1, v * g.y);
    unsafeAtomicAdd(dst + 2, v * g.z);
    unsafeAtomicAdd(dst + 3, v * g.w);
}

// ---------------------------------------------------------------------------
// adj_rec[i][j] = sigmoid(dot(z[i,:], z[j,:])), K = 64 -> 16 WMMA steps.
// B = z^T, so the B operand load is symmetric with A (both read z rows,
// one contiguous v2f per lane per step; z is 2MB -> L2-hot).
// One wave per 16x16 output tile; each result row is stored as two
// coalesced 64B segments with a NON-TEMPORAL hint: adj_rec is write-only
// and 268MB, so RT stores would thrash the entire 192MB L2.
// ---------------------------------------------------------------------------
__global__ void decoder_sigmoid_wmma(const float* __restrict__ z,
                                     float* __restrict__ out) {
    const int lane = threadIdx.x & 31;
    const int wave = (blockIdx.x * blockDim.x + threadIdx.x) >> 5;
    const int tilesN = N_NODES / 16;   // 512
    const int tm = wave / tilesN;
    const int tn = wave % tilesN;
    const int l16 = lane & 15;
    const int hi  = lane >> 4;

    v8f acc = {0.f, 0.f, 0.f, 0.f, 0.f, 0.f, 0.f, 0.f};
    const float* aRow = z + (size_t)(tm * 16 + l16) * EMB + 2 * hi;
    const float* bRow = z + (size_t)(tn * 16 + l16) * EMB + 2 * hi;

#pragma unroll
    for (int k = 0; k < EMB; k += 4) {
        v2f a = *(const v2f*)(aRow + k);
        v2f b = *(const v2f*)(bRow + k);
        acc = __builtin_amdgcn_wmma_f32_16x16x4_f32(
            false, a, false, b, (short)0, acc, false, false);
    }

    float* o = out + (size_t)(tm * 16 + 8 * hi) * N_NODES + tn * 16 + l16;
#pragma unroll
    for (int r = 0; r < 8; ++r) {
        float s = acc[r];
        float sig = 1.0f / (1.0f + __expf(-s));
        __builtin_nontemporal_store(sig, o + (size_t)r * N_NODES);
    }
}

extern "C" void kernel_launch(void* const* d_in, const int* in_sizes, int n_in,
                              void* d_out, int out_size, void* d_ws, size_t ws_size,
                              hipStream_t stream) {
    const float* x    = (const float*)d_in[0];   // [N, F_IN]
    const int*   rows = (const int*)  d_in[1];   // [E]
    const int*   cols = (const int*)  d_in[2];   // [E]
    const float* vals = (const float*)d_in[3];   // [E]
    const float* W1   = (const float*)d_in[4];   // [F_IN, HID]
    const float* b1   = (const float*)d_in[5];   // [HID]
    const float* W2   = (const float*)d_in[6];   // [HID, EMB]
    const float* b2   = (const float*)d_in[7];   // [EMB]

    float* z       = (float*)d_out;                         // [N, EMB]
    float* adj_rec = z + (size_t)N_NODES * EMB;             // [N, N]

    float* h1 = (float*)d_ws;                               // [N, HID]  8 MB
    float* h  = h1 + (size_t)N_NODES * HID;                 // [N, HID]  8 MB
    float* z1 = h  + (size_t)N_NODES * HID;                 // [N, EMB]  2 MB

    // Zero the atomic-scatter targets (d_ws/d_out are poisoned, not zeroed).
    zero_f32<<<(N_NODES * HID + 255) / 256, 256, 0, stream>>>(h, N_NODES * HID);
    zero_f32<<<(N_NODES * EMB + 255) / 256, 256, 0, stream>>>(z, N_NODES * EMB);

    // GEMM1: h1 = x @ W1 + b1    (8192/16 * 256/16 = 8192 waves)
    {
        int waves = (N_NODES / 16) * (HID / 16);
        gemm_bias_wmma<F_IN, HID, false>
            <<<waves * 32 / 256, 256, 0, stream>>>(x, W1, b1, h1);
    }
    // SpMM1: h = scatter(adj, h1)
    {
        int threads = N_EDGE * (HID / 4);
        spmm_scatter<HID><<<threads / 256, 256, 0, stream>>>(rows, cols, vals, h1, h);
    }
    // GEMM2: z1 = relu(h) @ W2 + b2   (512 * 4 = 2048 waves)
    {
        int waves = (N_NODES / 16) * (EMB / 16);
        gemm_bias_wmma<HID, EMB, true>
            <<<waves * 32 / 256, 256, 0, stream>>>(h, W2, b2, z1);
    }
    // SpMM2: z = scatter(adj, z1)
    {
        int threads = N_EDGE * (EMB / 4);
        spmm_scatter<EMB><<<threads / 256, 256, 0, stream>>>(rows, cols, vals, z1, z);
    }
    // Decoder: adj_rec = sigmoid(z @ z^T)   (512*512 = 262144 waves)
    {
        int waves = (N_NODES / 16) * (N_NODES / 16);
        decoder_sigmoid_wmma<<<waves * 32 / 256, 256, 0, stream>>>(z, adj_rec);
    }
}